// BinarizeLayer_38242388803637
// MI455X (gfx1250) — compile-verified
//
#include <hip/hip_runtime.h>
#include <stdint.h>

// out[b,f] = (medians[f] > 0) && (inputs[b,f] >= medians[f])  -> f32 0/1
// 16384 x 8192 f32 in, 8192 f32 medians, f32 out. Bandwidth-bound:
// ~1.07 GB HBM traffic -> ~46 us floor at 23.3 TB/s. No matmul structure,
// so no WMMA. CDNA5 features: TDM tensor_load_to_lds + s_wait_tensorcnt to
// stage the per-block 4KB median slice into LDS; NT-hinted B128 streaming.

typedef float float4v __attribute__((ext_vector_type(4)));
typedef int   v4i     __attribute__((ext_vector_type(4)));
typedef int   v8i     __attribute__((ext_vector_type(8)));

#define FEATURES 8192          // medians length (power of two)
#define BLOCKS   8192          // stride = BLOCKS*256 f4 = multiple of 2048
#define THREADS  256

__global__ __launch_bounds__(THREADS)
void binarize_tdm_kernel(const float* __restrict__ in,
                         const float* __restrict__ med,
                         float* __restrict__ out,
                         unsigned total4) {
  // 4 KB LDS slice: medians for float4-groups [(b%8)*256, (b%8)*256+256)
  __shared__ float smed[1024];

  // Low 32 bits of the flat LDS pointer == LDS byte offset (aperture in high bits).
  unsigned lds_base = (unsigned)(uintptr_t)(&smed[0]);

  unsigned slice = (blockIdx.x & 7u);
  uint64_t gaddr = (uint64_t)(uintptr_t)med + (uint64_t)slice * 4096ull;

  if ((threadIdx.x >> 5) == 0) {      // wave 0 issues the TDM copy
    v4i g0; v8i g1;
    // ---- D# group 0 (128b) ----
    g0[0] = 1;                                        // count=1 (valid), user mode
    g0[1] = (int)lds_base;                            // lds_addr (bytes)
    g0[2] = (int)(uint32_t)gaddr;                     // global_addr[31:0]
    g0[3] = (int)((uint32_t)(gaddr >> 32) | 0x80000000u); // addr[56:32] | type=2
    // ---- D# group 1 (256b) ----
    g1[0] = 0x00020000;                               // data_size=2 (4 bytes)
    g1[1] = (int)((FEATURES & 0xFFFFu) << 16);        // tensor_dim0 lo16 (8192)
    g1[2] = (int)((1u << 16) | ((FEATURES >> 16) & 0xFFFFu)); // dim0 hi16, tensor_dim1=1
    g1[3] = (int)(1024u << 16);                       // tile_dim0 = 1024 elements (4KB)
    g1[4] = 1;                                        // tile_dim1 = 1
    g1[5] = FEATURES;                                 // tensor_dim0_stride
    g1[6] = 0;
    g1[7] = 0;
    asm volatile("tensor_load_to_lds %0, %1" :: "s"(g0), "s"(g1) : "memory");
    __builtin_amdgcn_s_wait_tensorcnt(0);
  }
  __syncthreads();

  const float4v* __restrict__ in4  = (const float4v*)in;
  float4v*       __restrict__ out4 = (float4v*)out;

  // Thread t in block b handles f4 index (b*256+t) mod 2048 == slice*256 + t,
  // and the grid stride (8192*256) is a multiple of 2048, so the median float4
  // is loop-invariant: read it from LDS once, and fold the sign test into it.
  const float4v m = ((const float4v*)smed)[threadIdx.x];
  const float INF = __builtin_inff();
  float4v msel;                       // (m > 0) ? m : +INF  => single cmp in loop
  msel.x = (m.x > 0.0f) ? m.x : INF;
  msel.y = (m.y > 0.0f) ? m.y : INF;
  msel.z = (m.z > 0.0f) ? m.z : INF;
  msel.w = (m.w > 0.0f) ? m.w : INF;

  // 32-bit indexing: total4 = 33.5M < 2^31, byte extent 512 MiB < 2^31,
  // so the backend can use SADDR-base + 32-bit voffset addressing.
  const unsigned stride = (unsigned)gridDim.x * THREADS;
  unsigned i = blockIdx.x * THREADS + threadIdx.x;
#pragma unroll 4
  for (; i < total4; i += stride) {
    float4v x = __builtin_nontemporal_load(in4 + i);
    float4v r;
    r.x = (x.x >= msel.x) ? 1.0f : 0.0f;
    r.y = (x.y >= msel.y) ? 1.0f : 0.0f;
    r.z = (x.z >= msel.z) ? 1.0f : 0.0f;
    r.w = (x.w >= msel.w) ? 1.0f : 0.0f;
    __builtin_nontemporal_store(r, out4 + i);
  }
}

extern "C" void kernel_launch(void* const* d_in, const int* in_sizes, int n_in,
                              void* d_out, int out_size, void* d_ws, size_t ws_size,
                              hipStream_t stream) {
  const float* in  = (const float*)d_in[0];   // (16384, 8192) f32
  const float* med = (const float*)d_in[1];   // (8192,) f32
  float* out = (float*)d_out;                 // (16384, 8192) f32 0/1

  unsigned total  = (unsigned)in_sizes[0];    // 134,217,728
  unsigned total4 = total >> 2;

  binarize_tdm_kernel<<<dim3(BLOCKS), dim3(THREADS), 0, stream>>>(in, med, out, total4);
}